// GraphConvLayer_21363167330557
// MI455X (gfx1250) — compile-verified
//
#include <hip/hip_runtime.h>
#include <math.h>

// Problem constants (match reference).
#define NN     100000      // nodes
#define NE     1600000     // edges
#define DD     32          // feature dim
#define HH     32          // hidden dim
#define EPS_BN 1e-3f
#define EPS_L2 1e-12f

typedef float v2f __attribute__((ext_vector_type(2)));
typedef float v8f __attribute__((ext_vector_type(8)));

struct FfnP {
    const float *g1, *be1, *m1, *v1, *W1, *b1;   // BN1 (gamma,beta,mean,var), Dense1
    const float *g2, *be2, *m2, *v2, *W2, *b2;   // BN2, Dense2
};

__device__ __forceinline__ float gelu_f(float x) {
    // exact gelu: x * 0.5 * (1 + erf(x/sqrt(2)))
    return 0.5f * x * (1.0f + erff(x * 0.7071067811865475f));
}
__device__ __forceinline__ float bn_f(float x, const float* g, const float* b,
                                      const float* m, const float* v, int j) {
    return (x - m[j]) * rsqrtf(v[j] + EPS_BN) * g[j] + b[j];
}
__device__ __forceinline__ v8f wmma4(v2f a, v2f b, v8f c) {
    // V_WMMA_F32_16X16X4_F32: D = A(16x4) x B(4x16) + C, f32 throughout.
    return __builtin_amdgcn_wmma_f32_16x16x4_f32(false, a, false, b, (short)0, c,
                                                 false, false);
}

// ---------------------------------------------------------------------------
// Kernel 1: prep FFN over all nodes: P = gelu(BN2(gelu(BN1(X)@W1+b1))@W2+b2)
// One wave32 per 16-node tile; 8 waves / 256-thread block.
// ---------------------------------------------------------------------------
__global__ __launch_bounds__(256) void ffn_prep_kernel(
    const float* __restrict__ X, FfnP p, float* __restrict__ P)
{
    __shared__ float hidsm[8][16][HH + 1];       // per-wave transpose staging
    const int lane = threadIdx.x & 31;
    const int wave = threadIdx.x >> 5;
    const int l16  = lane & 15;
    const int half = lane >> 4;
    const int tile = blockIdx.x * 8 + wave;
    if (tile >= NN / 16) return;                 // whole-wave exit: EXEC stays all-1s
    const int r0 = tile * 16;

    const float* xrow = X + (size_t)(r0 + l16) * DD;

    // ---- layer 1: C = BN1(X_tile) @ W1, two 16x16 col tiles, K=32 ----
    v8f c0 = {}, c1 = {};
#pragma unroll
    for (int kc = 0; kc < DD / 4; ++kc) {
        const int k = kc * 4 + half * 2;
        v2f a, b0, b1;
        a.x = bn_f(xrow[k],     p.g1, p.be1, p.m1, p.v1, k);
        a.y = bn_f(xrow[k + 1], p.g1, p.be1, p.m1, p.v1, k + 1);
        b0.x = p.W1[k * HH + l16];            b0.y = p.W1[(k + 1) * HH + l16];
        b1.x = p.W1[k * HH + 16 + l16];       b1.y = p.W1[(k + 1) * HH + 16 + l16];
        c0 = wmma4(a, b0, c0);
        c1 = wmma4(a, b1, c1);
    }
    // epilogue: +bias, gelu, BN2; stage into LDS for layout transpose
#pragma unroll
    for (int i = 0; i < 8; ++i) {
        const int row = i + 8 * half;
        const int n0 = l16, n1 = 16 + l16;
        float h0 = gelu_f(c0[i] + p.b1[n0]);
        float h1 = gelu_f(c1[i] + p.b1[n1]);
        hidsm[wave][row][n0] = bn_f(h0, p.g2, p.be2, p.m2, p.v2, n0);
        hidsm[wave][row][n1] = bn_f(h1, p.g2, p.be2, p.m2, p.v2, n1);
    }
    __syncthreads();

    // ---- layer 2: D = H @ W2, K=32 ----
    v8f d0 = {}, d1 = {};
#pragma unroll
    for (int kc = 0; kc < HH / 4; ++kc) {
        const int k = kc * 4 + half * 2;
        v2f a, b0, b1;
        a.x = hidsm[wave][l16][k];
        a.y = hidsm[wave][l16][k + 1];
        b0.x = p.W2[k * HH + l16];            b0.y = p.W2[(k + 1) * HH + l16];
        b1.x = p.W2[k * HH + 16 + l16];       b1.y = p.W2[(k + 1) * HH + 16 + l16];
        d0 = wmma4(a, b0, d0);
        d1 = wmma4(a, b1, d1);
    }
#pragma unroll
    for (int i = 0; i < 8; ++i) {
        const int row = r0 + i + 8 * half;
        P[(size_t)row * HH + l16]      = gelu_f(d0[i] + p.b2[l16]);
        P[(size_t)row * HH + 16 + l16] = gelu_f(d1[i] + p.b2[16 + l16]);
    }
}

// ---------------------------------------------------------------------------
// Kernel 2: edge scatter. One wave32 per edge (lane = feature), 4 edges/wave.
// msgs = P[src] * w  ->  atomic add into sums[dst]; count per edge.
// ---------------------------------------------------------------------------
#define EDGES_PER_WAVE 4
__global__ __launch_bounds__(256) void edge_scatter_kernel(
    const int* __restrict__ edges,     // [2, E] (row0 = dst, row1 = src)
    const float* __restrict__ ew,      // [E]
    const float* __restrict__ P,       // [N, 32]
    float* __restrict__ sums,          // [N, 32]
    float* __restrict__ counts)        // [N]
{
    const int lane = threadIdx.x & 31;
    const int wave = threadIdx.x >> 5;
    const int base = (blockIdx.x * 8 + wave) * EDGES_PER_WAVE;
#pragma unroll
    for (int i = 0; i < EDGES_PER_WAVE; ++i) {
        const int e = base + i;
        const int dst = edges[e];
        const int src = edges[NE + e];
        const float w = ew[e];
        const float m = P[(size_t)src * HH + lane] * w;   // coalesced 128B row
        atomicAdd(&sums[(size_t)dst * HH + lane], m);     // global_atomic_add_f32
        if (lane == 0) atomicAdd(&counts[dst], 1.0f);
    }
}

// ---------------------------------------------------------------------------
// Kernel 3: update FFN + L2 norm.
// x = concat(features, sums/max(counts,1)) [N,64]; two dense layers; normalize.
// ---------------------------------------------------------------------------
__global__ __launch_bounds__(256) void ffn_update_kernel(
    const float* __restrict__ X,
    const float* __restrict__ sums,
    const float* __restrict__ counts,
    FfnP p, float* __restrict__ OUT)
{
    __shared__ float hidsm[8][16][HH + 1];
    const int lane = threadIdx.x & 31;
    const int wave = threadIdx.x >> 5;
    const int l16  = lane & 15;
    const int half = lane >> 4;
    const int tile = blockIdx.x * 8 + wave;
    if (tile >= NN / 16) return;
    const int r0 = tile * 16;

    const float* xrow = X    + (size_t)(r0 + l16) * DD;
    const float* srow = sums + (size_t)(r0 + l16) * HH;
    const float rinv  = 1.0f / fmaxf(counts[r0 + l16], 1.0f);

    // ---- layer 1: K = 64 (cols 0..31 features, 32..63 mean-agg) ----
    v8f c0 = {}, c1 = {};
#pragma unroll
    for (int kc = 0; kc < 16; ++kc) {
        const int k = kc * 4 + half * 2;
        const float x0 = (kc < 8) ? xrow[k]     : srow[k - 32] * rinv;
        const float x1 = (kc < 8) ? xrow[k + 1] : srow[k - 31] * rinv;
        v2f a, b0, b1;
        a.x = bn_f(x0, p.g1, p.be1, p.m1, p.v1, k);
        a.y = bn_f(x1, p.g1, p.be1, p.m1, p.v1, k + 1);
        b0.x = p.W1[k * HH + l16];            b0.y = p.W1[(k + 1) * HH + l16];
        b1.x = p.W1[k * HH + 16 + l16];       b1.y = p.W1[(k + 1) * HH + 16 + l16];
        c0 = wmma4(a, b0, c0);
        c1 = wmma4(a, b1, c1);
    }
#pragma unroll
    for (int i = 0; i < 8; ++i) {
        const int row = i + 8 * half;
        const int n0 = l16, n1 = 16 + l16;
        float h0 = gelu_f(c0[i] + p.b1[n0]);
        float h1 = gelu_f(c1[i] + p.b1[n1]);
        hidsm[wave][row][n0] = bn_f(h0, p.g2, p.be2, p.m2, p.v2, n0);
        hidsm[wave][row][n1] = bn_f(h1, p.g2, p.be2, p.m2, p.v2, n1);
    }
    __syncthreads();

    // ---- layer 2: K = 32 ----
    v8f d0 = {}, d1 = {};
#pragma unroll
    for (int kc = 0; kc < HH / 4; ++kc) {
        const int k = kc * 4 + half * 2;
        v2f a, b0, b1;
        a.x = hidsm[wave][l16][k];
        a.y = hidsm[wave][l16][k + 1];
        b0.x = p.W2[k * HH + l16];            b0.y = p.W2[(k + 1) * HH + l16];
        b1.x = p.W2[k * HH + 16 + l16];       b1.y = p.W2[(k + 1) * HH + 16 + l16];
        d0 = wmma4(a, b0, d0);
        d1 = wmma4(a, b1, d1);
    }

    // ---- epilogue: bias + gelu + row-wise L2 normalize (shfl reduction) ----
#pragma unroll
    for (int i = 0; i < 8; ++i) {
        float h0 = gelu_f(d0[i] + p.b2[l16]);
        float h1 = gelu_f(d1[i] + p.b2[16 + l16]);
        float ss = h0 * h0 + h1 * h1;
        ss += __shfl_xor(ss, 1, 32);   // xor<16 keeps reduction within each
        ss += __shfl_xor(ss, 2, 32);   // 16-lane half: exactly one output row
        ss += __shfl_xor(ss, 4, 32);
        ss += __shfl_xor(ss, 8, 32);
        const float sc = rsqrtf(fmaxf(ss, EPS_L2));
        const int row = r0 + i + 8 * half;
        OUT[(size_t)row * HH + l16]      = h0 * sc;
        OUT[(size_t)row * HH + 16 + l16] = h1 * sc;
    }
}

// ---------------------------------------------------------------------------
extern "C" void kernel_launch(void* const* d_in, const int* in_sizes, int n_in,
                              void* d_out, int out_size, void* d_ws, size_t ws_size,
                              hipStream_t stream) {
    (void)in_sizes; (void)n_in; (void)out_size; (void)ws_size;

    const float* X     = (const float*)d_in[0];   // features [N,32]
    const int*   edges = (const int*)d_in[1];     // [2,E]
    const float* ew    = (const float*)d_in[2];   // [E]

    // params flattened in setup_inputs insertion order:
    // prep: bn1(g,b,m,v), w1, b1, bn2(g,b,m,v), w2, b2 ; then upd: same.
    int i = 3;
    FfnP prep, upd;
    prep.g1 = (const float*)d_in[i++]; prep.be1 = (const float*)d_in[i++];
    prep.m1 = (const float*)d_in[i++]; prep.v1  = (const float*)d_in[i++];
    prep.W1 = (const float*)d_in[i++]; prep.b1  = (const float*)d_in[i++];
    prep.g2 = (const float*)d_in[i++]; prep.be2 = (const float*)d_in[i++];
    prep.m2 = (const float*)d_in[i++]; prep.v2  = (const float*)d_in[i++];
    prep.W2 = (const float*)d_in[i++]; prep.b2  = (const float*)d_in[i++];
    upd.g1  = (const float*)d_in[i++]; upd.be1  = (const float*)d_in[i++];
    upd.m1  = (const float*)d_in[i++]; upd.v1   = (const float*)d_in[i++];
    upd.W1  = (const float*)d_in[i++]; upd.b1   = (const float*)d_in[i++];
    upd.g2  = (const float*)d_in[i++]; upd.be2  = (const float*)d_in[i++];
    upd.m2  = (const float*)d_in[i++]; upd.v2   = (const float*)d_in[i++];
    upd.W2  = (const float*)d_in[i++]; upd.b2   = (const float*)d_in[i++];

    // Workspace: prep P [N,32] | sums [N,32] | counts [N]
    float* P      = (float*)d_ws;
    float* sums   = P + (size_t)NN * HH;
    float* counts = sums + (size_t)NN * HH;
    hipMemsetAsync(sums, 0, ((size_t)NN * HH + NN) * sizeof(float), stream);

    const int ntiles = NN / 16;                       // 6250 (exact)
    const int nblocks = (ntiles + 7) / 8;             // 782

    ffn_prep_kernel<<<nblocks, 256, 0, stream>>>(X, prep, P);

    const int eblocks = NE / (8 * EDGES_PER_WAVE);    // 50000 (exact)
    edge_scatter_kernel<<<eblocks, 256, 0, stream>>>(edges, ew, P, sums, counts);

    ffn_update_kernel<<<nblocks, 256, 0, stream>>>(X, sums, counts, upd,
                                                   (float*)d_out);
}